// LinearFieldSymmetricSourceBlock_13872744366310
// MI455X (gfx1250) — compile-verified
//
#include <hip/hip_runtime.h>
#include <hip/hip_bf16.h>
#include <math.h>

// ---------------- problem constants ----------------
#define NNODES 10000
#define NEDGES 160000
#define MULN   64
#define MULF   16
#define OFF_P  (NNODES * 4)   // p_ji output offset (after mpoles 10000x4)

#define LD 68                 // padded LDS stride (64 + 4) -> conflict-free A reads

// fast silu * gain: one v_exp + one v_rcp instead of IEEE divide sequence
static __device__ __forceinline__ float silu_gain(float z) {
    return 1.679177f * z * __builtin_amdgcn_rcpf(1.0f + __expf(-z));
}

typedef float v2f __attribute__((ext_vector_type(2)));
typedef float v8f __attribute__((ext_vector_type(8)));

// packed 12-byte triplet (4-byte aligned) -> lets backend emit global_load_b96
struct f3 { float x, y, z; };

// scale constants
#define INV_SQRT8 0.3535533905932738f
#define INV_SQ3   0.5773502691896258f
#define C1        0.00390625f              // alpha * 1/sqrt(128) = 1/256
#define C1V       (C1 * INV_SQ3)
#define A2C       0.022097086912079608f    // 1/sqrt(2048)

// ---------------------------------------------------------------------------
// Wave-level f32 WMMA GEMM tile: one wave computes 16 rows (edges) x 64 cols.
//   src: LDS activations [16][LD] (K features per row)
//   dst: LDS output      [16][LD] (64 cols)
//   W:   global weights, row-major [K][ldW], column base wcol0
// A layout (16x4 f32): lane L holds row M=L&15, VGPR pair = K {0,1} (lanes 0-15)
//   or K {2,3} (lanes 16-31).  B (4x16): lane = col, same K split.
// C/D (16x16 f32): VGPR r -> (M=r, N=lane) lanes 0-15 ; (M=8+r, N=lane-16).
// ---------------------------------------------------------------------------
static __device__ __forceinline__ void mlp_layer(
    const float* __restrict__ src, float* __restrict__ dst,
    const float* __restrict__ W, int K, int ldW, int wcol0,
    float inv, bool do_silu, int lane)
{
    const int row  = lane & 15;
    const int kh   = (lane < 16) ? 0 : 2;
    const int coll = lane & 15;
    const int orow = (lane < 16) ? 0 : 8;
    for (int t = 0; t < 4; ++t) {
        v8f c = {};
        const int col = wcol0 + t * 16 + coll;
        for (int kk = 0; kk < K; kk += 4) {
            const int k0 = kk + kh;
            v2f a, b;
            a.x = src[row * LD + k0];
            a.y = src[row * LD + k0 + 1];
            b.x = W[k0 * ldW + col];
            b.y = W[(k0 + 1) * ldW + col];
            c = __builtin_amdgcn_wmma_f32_16x16x4_f32(
                    false, a, false, b, (short)0, c, false, false);
        }
        #pragma unroll
        for (int rj = 0; rj < 8; ++rj) {
            float z = c[rj] * inv;
            dst[(orow + rj) * LD + t * 16 + coll] = do_silu ? silu_gain(z) : z;
        }
    }
}

// ---------------------------------------------------------------------------
// Node kernel: per-node linears l1/l2, folded field matrices Q, multipoles.
// One wave per node; 4 waves / block.
// ---------------------------------------------------------------------------
__global__ __launch_bounds__(128) void node_kernel(
    const float* __restrict__ nf,   const float* __restrict__ ff,
    const float* __restrict__ W1s,  const float* __restrict__ W1v,
    const float* __restrict__ W2s,  const float* __restrict__ W2v,
    const float* __restrict__ Wvs_s, const float* __restrict__ Wvs_v,
    const float* __restrict__ Wvr_s, const float* __restrict__ Wvr_v,
    const float* __restrict__ Wfs0, const float* __restrict__ Wfs1,
    const float* __restrict__ Wfr0, const float* __restrict__ Wfr1,
    const float* __restrict__ Wmp_sv, const float* __restrict__ Wmp_vs,
    float* __restrict__ l1s, float* __restrict__ l1v,
    float* __restrict__ l2s, float* __restrict__ l2v,
    float* __restrict__ Qss, float* __restrict__ Qvs,
    float* __restrict__ Qsr, float* __restrict__ Qvr,
    float* __restrict__ out)
{
    __shared__ float feats[4][256];
    __shared__ float fld[4][64];
    __shared__ float tb[4][128];

    const int wid  = threadIdx.x >> 5;
    const int lane = threadIdx.x & 31;
    const int n    = blockIdx.x * 4 + wid;

    float* F = feats[wid];
    float* G = fld[wid];
    float* T = tb[wid];

    #pragma unroll
    for (int q = 0; q < 8; ++q) F[lane * 8 + q] = nf[(size_t)n * 256 + lane * 8 + q];
    G[lane * 2 + 0] = ff[(size_t)n * 64 + lane * 2 + 0];
    G[lane * 2 + 1] = ff[(size_t)n * 64 + lane * 2 + 1];
    __syncthreads();

    // t vectors: [0:16)=Wfs0@fs  [16:64)=Wfs1@fv(:,i)  [64:80)=Wfr0@fs  [80:128)=Wfr1@fv
    for (int q = 0; q < 4; ++q) {
        const int idx = lane * 4 + q;
        float acc = 0.f;
        if (idx < 16) {
            const int j = idx;
            for (int v = 0; v < 16; ++v) acc += Wfs0[j * 16 + v] * G[v];
        } else if (idx < 64) {
            const int jj = idx - 16; const int j = jj / 3, i = jj % 3;
            for (int v = 0; v < 16; ++v) acc += Wfs1[j * 16 + v] * G[16 + v * 3 + i];
        } else if (idx < 80) {
            const int j = idx - 64;
            for (int v = 0; v < 16; ++v) acc += Wfr0[j * 16 + v] * G[v];
        } else {
            const int jj = idx - 80; const int j = jj / 3, i = jj % 3;
            for (int v = 0; v < 16; ++v) acc += Wfr1[j * 16 + v] * G[16 + v * 3 + i];
        }
        T[idx] = acc;
    }
    __syncthreads();

    // l1 / l2 irrep linears (inv = 1/8)
    for (int q = 0; q < 16; ++q) {
        const int idx = lane + q * 32;
        float acc = 0.f;
        if (idx < 64) {
            const int v = idx;
            for (int u = 0; u < 64; ++u) acc += F[u] * W1s[u * 64 + v];
            l1s[(size_t)n * 64 + v] = acc * 0.125f;
        } else if (idx < 256) {
            const int vv = idx - 64; const int v = vv / 3, i = vv % 3;
            for (int u = 0; u < 64; ++u) acc += F[64 + u * 3 + i] * W1v[u * 64 + v];
            l1v[(size_t)n * 192 + v * 3 + i] = acc * 0.125f;
        } else if (idx < 320) {
            const int v = idx - 256;
            for (int u = 0; u < 64; ++u) acc += F[u] * W2s[u * 64 + v];
            l2s[(size_t)n * 64 + v] = acc * 0.125f;
        } else {
            const int vv = idx - 320; const int v = vv / 3, i = vv % 3;
            for (int u = 0; u < 64; ++u) acc += F[64 + u * 3 + i] * W2v[u * 64 + v];
            l2v[(size_t)n * 192 + v * 3 + i] = acc * 0.125f;
        }
    }

    // folded field vectors Q (128 scalar + 128x3 vector, sender & receiver paths)
    for (int q = 0; q < 32; ++q) {
        const int idx = lane + q * 32;
        float acc = 0.f;
        if (idx < 128) {
            const int k = idx;
            for (int j = 0; j < 16; ++j) acc += Wvs_s[k * 16 + j] * T[j];
            Qss[(size_t)n * 128 + k] = acc * C1;
        } else if (idx < 512) {
            const int kk = idx - 128; const int k = kk / 3, i = kk % 3;
            for (int j = 0; j < 16; ++j) acc += Wvs_v[k * 16 + j] * T[16 + j * 3 + i];
            Qvs[(size_t)n * 384 + k * 3 + i] = acc * C1V;
        } else if (idx < 640) {
            const int k = idx - 512;
            for (int j = 0; j < 16; ++j) acc += Wvr_s[k * 16 + j] * T[64 + j];
            Qsr[(size_t)n * 128 + k] = acc * C1;
        } else {
            const int kk = idx - 640; const int k = kk / 3, i = kk % 3;
            for (int j = 0; j < 16; ++j) acc += Wvr_v[k * 16 + j] * T[80 + j * 3 + i];
            Qvr[(size_t)n * 384 + k * 3 + i] = acc * C1V;
        }
    }

    // multipoles mp1 (channel 0 is overwritten by charges -> write 0, edge atomics add)
    float mp0 = 0.f, mp1 = 0.f, mp2 = 0.f;
    #pragma unroll
    for (int q = 0; q < 2; ++q) {
        const int u = lane + q * 32;
        float gvs = 0.f;
        for (int v = 0; v < 16; ++v) gvs += Wmp_vs[u * 16 + v] * G[v];
        float gs0 = 0.f, gs1 = 0.f, gs2 = 0.f;
        for (int v = 0; v < 16; ++v) {
            const float w = Wmp_sv[u * 16 + v];
            gs0 += w * G[16 + v * 3 + 0];
            gs1 += w * G[16 + v * 3 + 1];
            gs2 += w * G[16 + v * 3 + 2];
        }
        mp0 += F[u] * gs0 + F[64 + u * 3 + 0] * gvs;
        mp1 += F[u] * gs1 + F[64 + u * 3 + 1] * gvs;
        mp2 += F[u] * gs2 + F[64 + u * 3 + 2] * gvs;
    }
    #pragma unroll
    for (int off = 16; off >= 1; off >>= 1) {
        mp0 += __shfl_xor(mp0, off, 32);
        mp1 += __shfl_xor(mp1, off, 32);
        mp2 += __shfl_xor(mp2, off, 32);
    }
    if (lane == 0) {
        out[n * 4 + 0] = 0.f;                    // charges accumulated by edge kernel
        out[n * 4 + 1] = 0.01f * A2C * mp0;
        out[n * 4 + 2] = 0.01f * A2C * mp1;
        out[n * 4 + 3] = 0.01f * A2C * mp2;
    }
}

// ---------------------------------------------------------------------------
// Edge kernel: 16 edges per wave. WMMA f32 MLP through LDS, then tensor-product
// + folded field dots chunk by chunk, shuffle-combine, scatter charges.
// ---------------------------------------------------------------------------
__global__ __launch_bounds__(128) void edge_kernel(
    const float* __restrict__ ef,   const float* __restrict__ eattr,
    const int*   __restrict__ eidx,
    const float* __restrict__ w1, const float* __restrict__ w2,
    const float* __restrict__ w3, const float* __restrict__ w4,
    const float* __restrict__ l1s, const float* __restrict__ l1v,
    const float* __restrict__ l2s, const float* __restrict__ l2v,
    const float* __restrict__ Qss, const float* __restrict__ Qvs,
    const float* __restrict__ Qsr, const float* __restrict__ Qvr,
    float* __restrict__ out)
{
    __shared__ float bufA[4][16 * LD];
    __shared__ float bufB[4][16 * LD];

    const int wid  = threadIdx.x >> 5;
    const int lane = threadIdx.x & 31;
    float* actA = bufA[wid];
    float* actB = bufB[wid];

    const int tile = blockIdx.x * 4 + wid;
    const int e0   = tile * 16;

    // stage edge_feats (16 edges x 8) into actA
    #pragma unroll
    for (int t = lane; t < 128; t += 32) {
        const int ed = t >> 3, c = t & 7;
        actA[ed * LD + c] = ef[(size_t)(e0 + ed) * 8 + c];
    }
    __syncthreads();

    mlp_layer(actA, actB, w1, 8, 64, 0, INV_SQRT8, true, lane);   // h1
    __syncthreads();
    mlp_layer(actB, actA, w2, 64, 64, 0, 0.125f, true, lane);     // h2
    __syncthreads();
    mlp_layer(actA, actB, w3, 64, 64, 0, 0.125f, true, lane);     // h3 (persists in actB)
    __syncthreads();

    // per-edge setup: 2 lanes per edge (halves of u-dimension)
    const int el = lane & 15;
    const int hh = lane >> 4;
    const int e  = e0 + el;
    const int s  = eidx[e];
    const int r  = eidx[NEDGES + e];
    const float y0  = eattr[(size_t)e * 4 + 0];
    const float y1x = eattr[(size_t)e * 4 + 1];
    const float y1y = eattr[(size_t)e * 4 + 2];
    const float y1z = eattr[(size_t)e * 4 + 3];

    const float* pl1s = l1s + (size_t)s * 64;
    const float* pl2s = l2s + (size_t)r * 64;
    const float* pl1v = l1v + (size_t)s * 192;
    const float* pl2v = l2v + (size_t)r * 192;
    const float* pQss = Qss + (size_t)s * 128;
    const float* pQsr = Qsr + (size_t)r * 128;
    const float* pQvs = Qvs + (size_t)s * 384;
    const float* pQvr = Qvr + (size_t)r * 384;

    float ps = 0.f, pr = 0.f;

    for (int j = 0; j < 4; ++j) {
        // tp_w chunk j (w[:,j,:], 16 edges x 64) -> actA
        mlp_layer(actB, actA, w4, 64, 256, j * 64, 0.125f, false, lane);
        __syncthreads();

        for (int uu = 0; uu < 32; ++uu) {
            const int u = hh * 32 + uu;
            const float wj = actA[el * LD + u];
            if (j == 0) {                 // m00 = w0 * bs * y0  (ms index u)
                const float bs = pl1s[u] + pl2s[u];
                const float m = wj * bs * y0;
                ps += m * pQss[u];
                pr += m * pQsr[u];
            } else if (j == 1) {          // m11 = w1 * (bv.y1)/sqrt3  (ms index 64+u)
                const f3 a = *(const f3*)(pl1v + u * 3);
                const f3 b = *(const f3*)(pl2v + u * 3);
                const float m = wj * ((a.x + b.x) * y1x + (a.y + b.y) * y1y +
                                      (a.z + b.z) * y1z) * INV_SQ3;
                ps += m * pQss[64 + u];
                pr += m * pQsr[64 + u];
            } else if (j == 2) {          // m01[u,i] = w2 * bs * y1[i]  (mv index u)
                const float bs = pl1s[u] + pl2s[u];
                const float ws_ = wj * bs;
                const f3 qs = *(const f3*)(pQvs + u * 3);
                const f3 qr = *(const f3*)(pQvr + u * 3);
                ps += ws_ * (y1x * qs.x + y1y * qs.y + y1z * qs.z);
                pr += ws_ * (y1x * qr.x + y1y * qr.y + y1z * qr.z);
            } else {                      // m10[u,i] = w3 * bv[u,i] * y0  (mv index 64+u)
                const f3 a = *(const f3*)(pl1v + u * 3);
                const f3 b = *(const f3*)(pl2v + u * 3);
                const float wy = wj * y0;
                const int k = 64 + u;
                const f3 qs = *(const f3*)(pQvs + k * 3);
                const f3 qr = *(const f3*)(pQvr + k * 3);
                ps += wy * ((a.x + b.x) * qs.x + (a.y + b.y) * qs.y + (a.z + b.z) * qs.z);
                pr += wy * ((a.x + b.x) * qr.x + (a.y + b.y) * qr.y + (a.z + b.z) * qr.z);
            }
        }
        __syncthreads();
    }

    // combine the two u-halves, emit p_ji and charge scatter
    ps += __shfl_xor(ps, 16, 32);
    pr += __shfl_xor(pr, 16, 32);
    if (lane < 16) {
        const float pji = 0.1f * (ps + pr);
        out[OFF_P + e] = pji;
        atomicAdd(out + (size_t)r * 4, pji);
        atomicAdd(out + (size_t)s * 4, -pji);
    }
}

// ---------------------------------------------------------------------------
extern "C" void kernel_launch(void* const* d_in, const int* in_sizes, int n_in,
                              void* d_out, int out_size, void* d_ws, size_t ws_size,
                              hipStream_t stream) {
    const float* node_feats = (const float*)d_in[1];
    const float* edge_attrs = (const float*)d_in[2];
    const float* edge_feats = (const float*)d_in[3];
    const float* field      = (const float*)d_in[4];
    const int*   eidx       = (const int*)d_in[5];
    const float* W1s  = (const float*)d_in[6];
    const float* W1v  = (const float*)d_in[7];
    const float* W2s  = (const float*)d_in[8];
    const float* W2v  = (const float*)d_in[9];
    const float* w1   = (const float*)d_in[10];
    const float* w2   = (const float*)d_in[11];
    const float* w3   = (const float*)d_in[12];
    const float* w4   = (const float*)d_in[13];
    const float* Wvs_s = (const float*)d_in[14];
    const float* Wvs_v = (const float*)d_in[15];
    const float* Wvr_s = (const float*)d_in[16];
    const float* Wvr_v = (const float*)d_in[17];
    const float* Wfs0 = (const float*)d_in[18];
    const float* Wfs1 = (const float*)d_in[19];
    const float* Wfr0 = (const float*)d_in[20];
    const float* Wfr1 = (const float*)d_in[21];
    // d_in[22] Wmp_ss, d_in[23] Wmp_vv: dead (mpoles[:,0] overwritten by charges)
    const float* Wmp_sv = (const float*)d_in[24];
    const float* Wmp_vs = (const float*)d_in[25];

    float* ws = (float*)d_ws;
    float* l1s = ws;                       // 10000*64
    float* l1v = l1s + 640000;             // 10000*192
    float* l2s = l1v + 1920000;
    float* l2v = l2s + 640000;
    float* Qss = l2v + 1920000;            // 10000*128
    float* Qvs = Qss + 1280000;            // 10000*384
    float* Qsr = Qvs + 3840000;
    float* Qvr = Qsr + 1280000;            // ends at 15,360,000 floats (61.4 MB)

    float* out = (float*)d_out;

    node_kernel<<<dim3(NNODES / 4), dim3(128), 0, stream>>>(
        node_feats, field, W1s, W1v, W2s, W2v,
        Wvs_s, Wvs_v, Wvr_s, Wvr_v, Wfs0, Wfs1, Wfr0, Wfr1,
        Wmp_sv, Wmp_vs,
        l1s, l1v, l2s, l2v, Qss, Qvs, Qsr, Qvr, out);

    edge_kernel<<<dim3(NEDGES / 16 / 4), dim3(128), 0, stream>>>(
        edge_feats, edge_attrs, eidx, w1, w2, w3, w4,
        l1s, l1v, l2s, l2v, Qss, Qvs, Qsr, Qvr, out);
}